// RPN2RoI_49185965473922
// MI455X (gfx1250) — compile-verified
//
#include <hip/hip_runtime.h>
#include <math.h>

#define B_IMGS   4
#define TOTK     4819            // 1000+1000+1000+1000+819
#define TILE     128
#define NCOLT    38              // ceil(4819/128)
#define MASKW    (NCOLT * 4)     // 152 u32 words per mask row
#define SORT_N   8192
#define MAXPER   1000
#define NMS_THR_F    0.7f
#define MAX_RATIO_F  4.135166556742356f

// Monotonic float -> uint key (larger float => larger key)
__device__ __forceinline__ unsigned fkey(float x) {
    unsigned u = __float_as_uint(x);
    return (u >> 31) ? ~u : (u | 0x80000000u);
}

// ---------------------------------------------------------------------------
// Kernel 1: exact top-k per (image, level) via 4-pass radix select on logits.
// cls layout: (B, A=3, H, W); flattened reference index j = (h*W+w)*3 + a.
// We scan in memory order m = a*HW + hw (coalesced) and remap to j on output.
// ---------------------------------------------------------------------------
__global__ __launch_bounds__(256) void topk_kernel(
    const float* __restrict__ cls, int N, int k, int lvloff,
    int* __restrict__ cand_idx, float* __restrict__ cand_score)
{
    const int b = blockIdx.x, tid = threadIdx.x, bs = blockDim.x;
    const int HW = N / 3;
    const float* __restrict__ p = cls + (size_t)b * N;

    __shared__ unsigned hist[256];
    __shared__ unsigned s_prefix, s_mask, s_kneed, s_cG, s_cT;
    if (tid == 0) { s_prefix = 0u; s_mask = 0u; s_kneed = (unsigned)k; }
    __syncthreads();

    for (int pass = 0; pass < 4; ++pass) {
        const int shift = 24 - 8 * pass;
        for (int i = tid; i < 256; i += bs) hist[i] = 0u;
        __syncthreads();
        const unsigned pre = s_prefix, msk = s_mask;
        for (int m = tid; m < N; m += bs) {
            if (m + 4 * bs < N) __builtin_prefetch(&p[m + 4 * bs], 0, 0);
            unsigned u = fkey(p[m]);
            if ((u & msk) == pre) atomicAdd(&hist[(u >> shift) & 255u], 1u);
        }
        __syncthreads();
        if (tid == 0) {
            unsigned need = s_kneed, cum = 0u;
            int bin = 255;
            for (; bin > 0; --bin) {
                unsigned c = hist[bin];
                if (cum + c >= need) break;
                cum += c;
            }
            s_kneed  = need - cum;
            s_prefix = pre | ((unsigned)bin << shift);
            s_mask   = msk | (255u << shift);
        }
        __syncthreads();
    }

    const unsigned T = s_prefix;
    const unsigned tieTake = s_kneed;
    const unsigned gCnt = (unsigned)k - tieTake;   // count of keys strictly > T
    if (tid == 0) { s_cG = 0u; s_cT = 0u; }
    __syncthreads();

    for (int m = tid; m < N; m += bs) {
        float x = p[m];
        unsigned u = fkey(x);
        int pos = -1;
        if (u > T) {
            pos = (int)atomicAdd(&s_cG, 1u);
        } else if (u == T) {
            unsigned t = atomicAdd(&s_cT, 1u);
            if (t < tieTake) pos = (int)(gCnt + t);
        }
        if (pos >= 0) {
            int j = (m % HW) * 3 + (m / HW);       // reference flat index
            size_t g = (size_t)b * TOTK + lvloff + pos;
            cand_idx[g]   = j;
            cand_score[g] = 1.0f / (1.0f + expf(-x));  // sigmoid applied late
        }
    }
}

// ---------------------------------------------------------------------------
// Kernel 2: decode + clip + area filter for all B*TOTK candidates.
// ---------------------------------------------------------------------------
__global__ __launch_bounds__(256) void decode_kernel(
    const float* r0, const float* r1, const float* r2, const float* r3, const float* r4,
    const float* a0, const float* a1, const float* a2, const float* a3, const float* a4,
    const int* __restrict__ imh, const int* __restrict__ imw,
    const int* __restrict__ cand_idx,
    float* __restrict__ cand_score, float* __restrict__ cand_box)
{
    int g = blockIdx.x * blockDim.x + threadIdx.x;
    if (g >= B_IMGS * TOTK) return;
    int b = g / TOTK, s = g % TOTK;
    int lvl = s / 1000; if (lvl > 4) lvl = 4;

    const float* regs[5] = { r0, r1, r2, r3, r4 };
    const float* ancs[5] = { a0, a1, a2, a3, a4 };
    const int    HWl[5]  = { 67200, 16800, 4200, 1050, 273 };

    int j  = cand_idx[g];
    int a  = j % 3, hw = j / 3;
    const float* rg = regs[lvl];
    size_t HW = (size_t)HWl[lvl];
    size_t rb = ((size_t)b * 12 + 4 * a) * HW + hw;   // reg: (B, 12, H*W)
    float d0 = rg[rb], d1 = rg[rb + HW], d2 = rg[rb + 2 * HW], d3 = rg[rb + 3 * HW];

    const float* an = ancs[lvl] + (size_t)j * 4;
    float x0 = an[0], y0 = an[1], x1 = an[2], y1 = an[3];
    float pw = x1 - x0, ph = y1 - y0;
    float px = x0 + 0.5f * pw, py = y0 + 0.5f * ph;
    float dw = fminf(d2, MAX_RATIO_F), dh = fminf(d3, MAX_RATIO_F);
    float gw = pw * expf(dw), gh = ph * expf(dh);
    float gx = px + pw * d0, gy = py + ph * d1;

    float fw = (float)(*imw), fh = (float)(*imh);
    float bx0 = fminf(fmaxf(gx - 0.5f * gw, 0.0f), fw);
    float by0 = fminf(fmaxf(gy - 0.5f * gh, 0.0f), fh);
    float bx1 = fminf(fmaxf(gx + 0.5f * gw, 0.0f), fw);
    float by1 = fminf(fmaxf(gy + 0.5f * gh, 0.0f), fh);

    cand_box[(size_t)g * 4 + 0] = bx0;
    cand_box[(size_t)g * 4 + 1] = by0;
    cand_box[(size_t)g * 4 + 2] = bx1;
    cand_box[(size_t)g * 4 + 3] = by1;

    float area = (bx1 - bx0) * (by1 - by0);
    if (!(area > 0.0f)) cand_score[g] = -1.0f;      // MIN_AREA = 0.0, strict >
}

// ---------------------------------------------------------------------------
// Kernel 3: per-image bitonic sort (descending by score) in LDS, then gather.
// 8192 padded elements, key = ~fkey(score) sorted ascending. 48KB LDS.
// ---------------------------------------------------------------------------
__global__ __launch_bounds__(1024) void sort_kernel(
    const float* __restrict__ cand_score, const float* __restrict__ cand_box,
    float* __restrict__ sscore, float* __restrict__ sboxes, int* __restrict__ slvl)
{
    const int b = blockIdx.x, tid = threadIdx.x, bs = blockDim.x;
    __shared__ unsigned       skey[SORT_N];
    __shared__ unsigned short sval[SORT_N];

    for (int i = tid; i < SORT_N; i += bs) {
        if (i < TOTK) {
            skey[i] = ~fkey(cand_score[(size_t)b * TOTK + i]);
            sval[i] = (unsigned short)i;
        } else {
            skey[i] = 0xFFFFFFFFu;
            sval[i] = 0xFFFFu;
        }
    }
    __syncthreads();

    for (unsigned k2 = 2; k2 <= SORT_N; k2 <<= 1) {
        for (unsigned j = k2 >> 1; j > 0; j >>= 1) {
            for (unsigned t = tid; t < SORT_N; t += bs) {
                unsigned ixj = t ^ j;
                if (ixj > t) {
                    bool asc = ((t & k2) == 0);
                    unsigned ka = skey[t], kb = skey[ixj];
                    if ((ka > kb) == asc) {
                        skey[t] = kb; skey[ixj] = ka;
                        unsigned short tv = sval[t]; sval[t] = sval[ixj]; sval[ixj] = tv;
                    }
                }
            }
            __syncthreads();
        }
    }

    for (int r = tid; r < TOTK; r += bs) {
        int slot = sval[r];
        int lv = slot / 1000; if (lv > 4) lv = 4;
        size_t src = (size_t)b * TOTK + slot;
        size_t dst = (size_t)b * TOTK + r;
        sscore[dst] = cand_score[src];
        slvl[dst]   = lv;
        sboxes[dst * 4 + 0] = cand_box[src * 4 + 0];
        sboxes[dst * 4 + 1] = cand_box[src * 4 + 1];
        sboxes[dst * 4 + 2] = cand_box[src * 4 + 2];
        sboxes[dst * 4 + 3] = cand_box[src * 4 + 3];
    }
}

// ---------------------------------------------------------------------------
// Kernel 4: suppression bitmask. Tile (row 128 x col 128). Column boxes are
// staged into LDS with the CDNA5 async-to-LDS engine (one B128 per lane,
// GVS addressing), waited with s_wait_asynccnt. Cross-level IoU is 0 by
// construction (lv*off offset trick in the reference), so compare levels.
// ---------------------------------------------------------------------------
__global__ __launch_bounds__(TILE) void ioumask_kernel(
    const float* __restrict__ sboxes, const int* __restrict__ slvl,
    unsigned* __restrict__ mask)
{
    const int bx = blockIdx.x, by = blockIdx.y, b = blockIdx.z, t = threadIdx.x;
    __shared__ float4 tb[TILE];
    __shared__ int    tl[TILE];

    int j0 = by * TILE + t;
    int jc = j0 < TOTK ? j0 : (TOTK - 1);           // clamp: duplicate, never read

    // Async stage of the 128 column boxes (16B each) into LDS.
    unsigned lds_addr = (unsigned)(size_t)(const void*)(&tb[t]);
    unsigned goff     = (unsigned)((unsigned)(b * TOTK + jc) * 16u);
    asm volatile("global_load_async_to_lds_b128 %0, %1, %2 offset:0"
                 :: "v"(lds_addr), "v"(goff), "s"(sboxes) : "memory");
    tl[t] = slvl[(size_t)b * TOTK + jc];
    asm volatile("s_wait_asynccnt 0x0" ::: "memory");
    __syncthreads();

    int i = bx * TILE + t;
    if (i >= TOTK) return;

    const float4 bi = reinterpret_cast<const float4*>(sboxes)[(size_t)b * TOTK + i];
    const int    li = slvl[(size_t)b * TOTK + i];
    const float  ai = (bi.z - bi.x) * (bi.w - bi.y);

    unsigned* mrow = mask + ((size_t)b * TOTK + i) * MASKW + (size_t)by * (TILE / 32);
    #pragma unroll
    for (int w = 0; w < TILE / 32; ++w) {
        unsigned bits = 0u;
        for (int c = 0; c < 32; ++c) {
            int j = by * TILE + w * 32 + c;
            if (j < TOTK && j > i && tl[w * 32 + c] == li) {
                float4 bj = tb[w * 32 + c];
                float xx0 = fmaxf(bi.x, bj.x), yy0 = fmaxf(bi.y, bj.y);
                float xx1 = fminf(bi.z, bj.z), yy1 = fminf(bi.w, bj.w);
                float iw = fmaxf(xx1 - xx0, 0.0f), ih = fmaxf(yy1 - yy0, 0.0f);
                float inter = iw * ih;
                float aj = (bj.z - bj.x) * (bj.w - bj.y);
                float uni = fmaxf(ai + aj - inter, 1e-6f);
                if (inter > NMS_THR_F * uni) bits |= (1u << c);
            }
        }
        mrow[w] = bits;
    }
}

// ---------------------------------------------------------------------------
// Kernel 5: greedy NMS reduction + compaction, one block per image.
// ---------------------------------------------------------------------------
__global__ __launch_bounds__(256) void nms_kernel(
    const float* __restrict__ sboxes, const float* __restrict__ sscore,
    const unsigned* __restrict__ mask, float* __restrict__ out)
{
    const int b = blockIdx.x, tid = threadIdx.x, bs = blockDim.x;
    __shared__ unsigned removed[MASKW];
    __shared__ int s_alive;

    for (int w = tid; w < MASKW; w += bs) removed[w] = 0u;
    __syncthreads();

    for (int i = 0; i < TOTK; ++i) {
        if (tid == 0) s_alive = !((removed[i >> 5] >> (i & 31)) & 1u);
        __syncthreads();
        if (s_alive) {
            const unsigned* mrow = mask + ((size_t)b * TOTK + i) * MASKW;
            for (int w = tid; w < MASKW; w += bs) removed[w] |= mrow[w];
        }
        __syncthreads();
    }

    if (tid == 0) {
        float* obox = out;                                 // (B, 1000, 4)
        float* osc  = out + (size_t)B_IMGS * MAXPER * 4;   // (B, 1000)
        int rank = 0;
        for (int i = 0; i < TOTK && rank < MAXPER; ++i) {
            bool kp = !((removed[i >> 5] >> (i & 31)) & 1u) &&
                      (sscore[(size_t)b * TOTK + i] >= 0.0f);
            if (kp) {
                size_t src = (size_t)b * TOTK + i;
                size_t dst = (size_t)b * MAXPER + rank;
                obox[dst * 4 + 0] = sboxes[src * 4 + 0];
                obox[dst * 4 + 1] = sboxes[src * 4 + 1];
                obox[dst * 4 + 2] = sboxes[src * 4 + 2];
                obox[dst * 4 + 3] = sboxes[src * 4 + 3];
                osc[dst] = sscore[src];
                ++rank;
            }
        }
    }
}

__global__ __launch_bounds__(256) void zero_kernel(float* __restrict__ out, int n)
{
    int g = blockIdx.x * blockDim.x + threadIdx.x;
    if (g < n) out[g] = 0.0f;
}

// ---------------------------------------------------------------------------
extern "C" void kernel_launch(void* const* d_in, const int* in_sizes, int n_in,
                              void* d_out, int out_size, void* d_ws, size_t ws_size,
                              hipStream_t stream)
{
    (void)in_sizes; (void)n_in; (void)out_size; (void)ws_size;

    // setup_inputs() dict order: cls0, reg0, anc0, cls1, reg1, anc1, ..., image_h, image_w
    const float* cls[5]; const float* reg[5]; const float* anc[5];
    for (int l = 0; l < 5; ++l) {
        cls[l] = (const float*)d_in[3 * l + 0];
        reg[l] = (const float*)d_in[3 * l + 1];
        anc[l] = (const float*)d_in[3 * l + 2];
    }
    const int* imh = (const int*)d_in[15];
    const int* imw = (const int*)d_in[16];

    // Workspace carve-out (all chunks are 16-byte multiples).
    char* w = (char*)d_ws;
    int*      cand_idx   = (int*)w;       w += (size_t)B_IMGS * TOTK * sizeof(int);
    float*    cand_score = (float*)w;     w += (size_t)B_IMGS * TOTK * sizeof(float);
    float*    cand_box   = (float*)w;     w += (size_t)B_IMGS * TOTK * 4 * sizeof(float);
    float*    sscore     = (float*)w;     w += (size_t)B_IMGS * TOTK * sizeof(float);
    float*    sboxes     = (float*)w;     w += (size_t)B_IMGS * TOTK * 4 * sizeof(float);
    int*      slvl       = (int*)w;       w += (size_t)B_IMGS * TOTK * sizeof(int);
    unsigned* mask       = (unsigned*)w;  // B*TOTK*MASKW u32 ≈ 11.7 MB

    static const int Nl[5] = { 201600, 50400, 12600, 3150, 819 };
    static const int Kl[5] = { 1000, 1000, 1000, 1000, 819 };
    static const int Ol[5] = { 0, 1000, 2000, 3000, 4000 };

    for (int l = 0; l < 5; ++l)
        topk_kernel<<<dim3(B_IMGS), dim3(256), 0, stream>>>(
            cls[l], Nl[l], Kl[l], Ol[l], cand_idx, cand_score);

    decode_kernel<<<dim3((B_IMGS * TOTK + 255) / 256), dim3(256), 0, stream>>>(
        reg[0], reg[1], reg[2], reg[3], reg[4],
        anc[0], anc[1], anc[2], anc[3], anc[4],
        imh, imw, cand_idx, cand_score, cand_box);

    sort_kernel<<<dim3(B_IMGS), dim3(1024), 0, stream>>>(
        cand_score, cand_box, sscore, sboxes, slvl);

    ioumask_kernel<<<dim3(NCOLT, NCOLT, B_IMGS), dim3(TILE), 0, stream>>>(
        sboxes, slvl, mask);

    zero_kernel<<<dim3((B_IMGS * MAXPER * 5 + 255) / 256), dim3(256), 0, stream>>>(
        (float*)d_out, B_IMGS * MAXPER * 5);

    nms_kernel<<<dim3(B_IMGS), dim3(256), 0, stream>>>(
        sboxes, sscore, mask, (float*)d_out);
}